// TeacherForcer_81338090651873
// MI455X (gfx1250) — compile-verified
//
#include <hip/hip_runtime.h>

#define NEGV (-1000000000.0f)

typedef float v2f __attribute__((ext_vector_type(2)));
typedef float v8f __attribute__((ext_vector_type(8)));

__device__ __constant__ float VALENCY_C[12] =
    {4.f, 1.f, 3.f, 1.f, 2.f, 1.f, 5.f, 1.f, 6.f, 1.f, 1.f, 1000.f};

__device__ inline void atomAddF(float* p, float v) { unsafeAtomicAdd(p, v); }

// ---------------------------------------------------------------- utilities
__global__ void fill_kernel(float* p, float v, long long n) {
  long long i = (long long)blockIdx.x * blockDim.x + threadIdx.x;
  long long stride = (long long)gridDim.x * blockDim.x;
  for (; i < n; i += stride) p[i] = v;
}

__global__ void writeout_kernel(float* out, const float* lp) { out[0] = lp[0]; }

__global__ void vecdiv_kernel(float* z, const float* sum, float invn) {
  z[threadIdx.x] = sum[threadIdx.x] * invn;
}

// ------------------------------------------------------ generic WMMA GEMM
// C[M x 128] = A[M x K] @ B[K x 128]; one wave computes a 16-row stripe.
// Software-pipelined: fragments for step k+1 are loaded while the 8 WMMAs of
// step k execute, so the wave keeps ~17 loads in flight instead of waiting
// on loadcnt==0 before every WMMA.
// f32 WMMA 16x16x4 fragment layouts per CDNA5 ISA 7.12.2:
//   A (16x4): lane<16 -> row=lane, holds K=0,1; lane>=16 -> row=lane-16, K=2,3
//   B (4x16): lane<16 -> col=lane, K=0,1; lane>=16 -> col=lane-16, K=2,3
//   C (16x16): vgpr e, lane<16 -> (M=e, N=lane); lane>=16 -> (M=e+8, N=lane-16)
__global__ __launch_bounds__(32)
void gemm_wmma_n128(const float* __restrict__ A, const float* __restrict__ B,
                    float* __restrict__ C, int M, int K,
                    long long aStride, long long cStride) {
  const int N = 128;
  A += (long long)blockIdx.y * aStride;
  C += (long long)blockIdx.y * cStride;
  int lane = threadIdx.x & 31;
  int rowTile = blockIdx.x << 4;
  int r = rowTile + (lane & 15);
  int kh = (lane >> 4) << 1;  // 0 or 2
  int nlane = lane & 15;

  // Out-of-range rows: clamp the address (stays in-bounds) and zero the frag.
  int rC = (r < M) ? r : (M - 1);
  float aScale = (r < M) ? 1.0f : 0.0f;
  const float* aRow = A + (long long)rC * K + kh;
  const float* bRow0 = B + (long long)kh * N + nlane;
  const float* bRow1 = bRow0 + N;

  v8f acc[8];
#pragma unroll
  for (int j = 0; j < 8; ++j)
#pragma unroll
    for (int e = 0; e < 8; ++e) acc[j][e] = 0.0f;

  v2f a_cur, a_nxt;
  v2f b_cur[8], b_nxt[8];

  // prologue: load fragments for k0 = 0
  a_cur.x = aRow[0] * aScale;
  a_cur.y = aRow[1] * aScale;
#pragma unroll
  for (int j = 0; j < 8; ++j) {
    b_cur[j].x = bRow0[j << 4];
    b_cur[j].y = bRow1[j << 4];
  }

  for (int k0 = 0; k0 < K; k0 += 4) {
    int kn = k0 + 4;
    if (kn < K) {  // prefetch next step's fragments
      a_nxt.x = aRow[kn] * aScale;
      a_nxt.y = aRow[kn + 1] * aScale;
      const float* b0 = bRow0 + (long long)kn * N;
      const float* b1 = bRow1 + (long long)kn * N;
#pragma unroll
      for (int j = 0; j < 8; ++j) {
        b_nxt[j].x = b0[j << 4];
        b_nxt[j].y = b1[j << 4];
      }
    }
#pragma unroll
    for (int j = 0; j < 8; ++j)
      acc[j] = __builtin_amdgcn_wmma_f32_16x16x4_f32(
          false, a_cur, false, b_cur[j], (short)0, acc[j], false, false);
    a_cur = a_nxt;
#pragma unroll
    for (int j = 0; j < 8; ++j) b_cur[j] = b_nxt[j];
  }

  int rowBase = rowTile + ((lane >> 4) << 3);
#pragma unroll
  for (int e = 0; e < 8; ++e) {
    int row = rowBase + e;
    if (row < M) {
#pragma unroll
      for (int j = 0; j < 8; ++j)
        C[(long long)row * N + (j << 4) + nlane] = acc[j][e];
    }
  }
}

// ------------------------------------------------------------- GCN kernels
__global__ __launch_bounds__(256)
void edge_deg_kernel(float* deg, const int* __restrict__ dst, int nE) {
  int e = blockIdx.x * blockDim.x + threadIdx.x;
  if (e < nE) atomAddF(&deg[dst[e]], 1.0f);
}

__global__ __launch_bounds__(256)
void edge_agg_kernel(float* __restrict__ agg, const float* __restrict__ h,
                     const float* __restrict__ deg, const int* __restrict__ src,
                     const int* __restrict__ dst, int nE) {
  long long gid = (long long)blockIdx.x * blockDim.x + threadIdx.x;
  int e = (int)(gid >> 5);
  int lane = (int)(gid & 31);
  if (e >= nE) return;
  int s = src[e], d = dst[e];
  float nrm = rsqrtf(deg[s] * deg[d]);
  const float4* hp =
      reinterpret_cast<const float4*>(h + (long long)s * 128 + (lane << 2));
  float4 hv = *hp;
  float* ap = agg + (long long)d * 128 + (lane << 2);
  atomAddF(ap + 0, hv.x * nrm);
  atomAddF(ap + 1, hv.y * nrm);
  atomAddF(ap + 2, hv.z * nrm);
  atomAddF(ap + 3, hv.w * nrm);
}

__global__ __launch_bounds__(256)
void combine_kernel(float* __restrict__ out, const float* __restrict__ agg,
                    const float* __restrict__ h, const float* __restrict__ deg,
                    const float* __restrict__ b, long long n, int relu) {
  long long gid = (long long)blockIdx.x * blockDim.x + threadIdx.x;
  if (gid >= n * 128) return;
  long long i = gid >> 7;
  int f = (int)(gid & 127);
  float v = agg[gid] + h[gid] / deg[i] + b[f];
  if (relu) v = fmaxf(v, 0.0f);
  out[gid] = v;
}

__global__ __launch_bounds__(256)
void rowmean_kernel(float* sum, const float* __restrict__ X, int n) {
  __shared__ float red[256];
  int f = threadIdx.x & 127, slice = threadIdx.x >> 7;
  float local = 0.0f;
  for (int i = blockIdx.x * 2 + slice; i < n; i += gridDim.x * 2)
    local += X[(long long)i * 128 + f];
  red[threadIdx.x] = local;
  __syncthreads();
  if (slice == 0) atomAddF(&sum[f], red[threadIdx.x] + red[threadIdx.x + 128]);
}

// ----------------------------------------------------- ligand label kernels
__global__ void xlabel_kernel(float* __restrict__ xlf,
                              const float* __restrict__ zlig,
                              const float* __restrict__ Wf,
                              const float* __restrict__ bf, int Nl, int A) {
  int i = blockIdx.x * blockDim.x + threadIdx.x;
  if (i > Nl) return;
  if (i == Nl) {  // appended one_hot(A-1) row
    for (int j = 0; j < A; ++j) xlf[(long long)i * A + j] = (j == A - 1) ? 1.0f : 0.0f;
    return;
  }
  float lg[16];
  float m = -3.0e38f;
  for (int j = 0; j < A; ++j) {
    float acc = bf[j];
    for (int k = 0; k < 128; ++k) acc += zlig[(long long)i * 128 + k] * Wf[k * A + j];
    lg[j] = acc;
    if (acc > m) m = acc;
  }
  float Z = 0.0f;
  for (int j = 0; j < A; ++j) { lg[j] = expf(lg[j] - m); Z += lg[j]; }
  for (int j = 0; j < A; ++j) xlf[(long long)i * A + j] = lg[j] / Z;
}

__global__ __launch_bounds__(512)
void lp0_kernel(float* lp, const float* __restrict__ xlf,
                const float* __restrict__ x_l, int Nl, int Fl, int A) {
  __shared__ float red[512];
  int i = threadIdx.x;
  float term = 0.0f;
  if (i < Nl) {
    float p = 0.0f;
    for (int j = 0; j < A; ++j) p += xlf[(long long)i * A + j] * x_l[(long long)i * Fl + 4 + j];
    term = logf(p);
  }
  red[i] = term;
  __syncthreads();
  for (int st = 256; st > 0; st >>= 1) {
    if (i < st) red[i] += red[i + st];
    __syncthreads();
  }
  if (i == 0) atomAddF(lp, red[0]);
}

// ------------------------------------------------ deterministic step state
__global__ __launch_bounds__(512)
void stepstate_kernel(int* us, int* vs, float* valS, float* degS,
                      const int* __restrict__ bfs_index,
                      const float* __restrict__ bfs_attr,
                      const float* __restrict__ x_l, int N1, int T, int Fl,
                      int A) {
  __shared__ int su, sv;
  __shared__ float sb;
  int i = threadIdx.x;
  float val = 0.0f, deg = 1.0f;
  if (i < N1) {
    if (i < N1 - 1) {
      int am = 0;
      float mv = x_l[(long long)i * Fl + 4];
      for (int j = 1; j < A; ++j) {
        float v = x_l[(long long)i * Fl + 4 + j];
        if (v > mv) { mv = v; am = j; }
      }
      if (am > 11) am = 11;
      val = VALENCY_C[am];
    } else {
      val = 100.0f;
    }
  }
  for (int t = 0; t < T; ++t) {
    if (i < N1) {
      valS[(long long)t * N1 + i] = val;
      degS[(long long)t * N1 + i] = deg;
    }
    if (i == 0) {
      int u = bfs_index[2 * t], v = bfs_index[2 * t + 1];
      us[t] = u; vs[t] = v; su = u; sv = v;
      int am = 0;
      float mv = bfs_attr[t * 3];
      for (int j = 1; j < 3; ++j) {
        float x = bfs_attr[t * 3 + j];
        if (x > mv) { mv = x; am = j; }
      }
      sb = 1.0f + (float)am;
    }
    __syncthreads();
    if (i == su || i == sv) { val -= sb; deg += 1.0f; }
    __syncthreads();
  }
}

// ---------------------------------------------- per-step GCN layer kernels
__global__ __launch_bounds__(128)
void z1scan_kernel(float* __restrict__ z1, const float* __restrict__ h1d,
                   const int* __restrict__ us, const int* __restrict__ vs,
                   const float* __restrict__ degS,
                   const float* __restrict__ b1, int N1) {
  const int D = 128;
  int t = blockIdx.y, i = blockIdx.x, f = threadIdx.x;
  float acc = 0.0f, degi;
  if (t == 0) {  // z0 graph: single directed edge u0->v0 over Nl nodes
    int u0 = us[0], v0 = vs[0];
    degi = (i == v0) ? 2.0f : 1.0f;
    if (i == v0) acc = h1d[(long long)u0 * D + f] * rsqrtf(2.0f);
  } else {
    degi = degS[(long long)t * N1 + i];
    for (int s = 0; s < t; ++s) {
      int u = us[s], v = vs[s];
      if (u == i || v == i) {
        float nrm = rsqrtf(degS[(long long)t * N1 + u] * degS[(long long)t * N1 + v]);
        if (v == i) acc += h1d[(long long)u * D + f] * nrm;
        if (u == i) acc += h1d[(long long)v * D + f] * nrm;
      }
    }
  }
  float val = acc + h1d[(long long)i * D + f] / degi + b1[f];
  float z = fmaxf(val, 0.0f);
  if (t == 0 && i == N1 - 1) z = 0.0f;
  z1[((long long)t * N1 + i) * D + f] = z;
}

__global__ __launch_bounds__(128)
void zvscan_kernel(float* __restrict__ zv, const float* __restrict__ h2,
                   const int* __restrict__ us, const int* __restrict__ vs,
                   const float* __restrict__ degS,
                   const float* __restrict__ b2, int N1) {
  const int D = 128;
  int t = blockIdx.y, i = blockIdx.x, f = threadIdx.x;
  const float* h2t = h2 + (long long)t * N1 * D;
  float acc = 0.0f, degi;
  if (t == 0) {
    int u0 = us[0], v0 = vs[0];
    degi = (i == v0) ? 2.0f : 1.0f;
    if (i == v0) acc = h2t[(long long)u0 * D + f] * rsqrtf(2.0f);
  } else {
    degi = degS[(long long)t * N1 + i];
    for (int s = 0; s < t; ++s) {
      int u = us[s], v = vs[s];
      if (u == i || v == i) {
        float nrm = rsqrtf(degS[(long long)t * N1 + u] * degS[(long long)t * N1 + v]);
        if (v == i) acc += h2t[(long long)u * D + f] * nrm;
        if (u == i) acc += h2t[(long long)v * D + f] * nrm;
      }
    }
  }
  float val = acc + h2t[(long long)i * D + f] / degi + b2[f];
  if (t == 0 && i == N1 - 1) val = 0.0f;  // z_v0 = concat(z0, zeros)
  zv[((long long)t * N1 + i) * D + f] = val;
}

__global__ __launch_bounds__(128)
void zg_kernel(float* __restrict__ zg, const float* __restrict__ zv, int N1) {
  const int D = 128;
  int t = blockIdx.x, f = threadIdx.x;
  float s = 0.0f;
  const float* p = zv + (long long)t * N1 * D + f;
  for (int i = 0; i < N1; ++i) s += p[(long long)i * D];
  float div = (t == 0) ? (float)(N1 - 1) : (float)N1;  // z_g0 is mean over Nl rows
  zg[(long long)t * D + f] = s / div;
}

// --------------------------------------------------------- per-step scoring
__global__ __launch_bounds__(512)
void score_kernel(float* __restrict__ lp_acc, const float* __restrict__ zv_all,
                  const float* __restrict__ zg_all,
                  const float* __restrict__ xlf,
                  const float* __restrict__ z_pocket,
                  const float* __restrict__ z_ligand,
                  const float* __restrict__ valS, const int* __restrict__ us,
                  const int* __restrict__ vs,
                  const float* __restrict__ bfs_attr,
                  const float* __restrict__ Wg, const float* __restrict__ bg,
                  const float* __restrict__ Wh, const float* __restrict__ bh,
                  int N1, int A) {
  const int D = 128;
  __shared__ float red[512];
  __shared__ float phiv[768];
  __shared__ int eu[512], ev[512];
  __shared__ float sh_c0, sh_sv, sh_lse, sh_lg[3];
  __shared__ int sh_u, sh_v;

  int t = blockIdx.x;
  int tid = threadIdx.x;

  for (int s = tid; s < t; s += 512) { eu[s] = us[s]; ev[s] = vs[s]; }
  if (tid == 0) { sh_u = us[t]; sh_v = vs[t]; }
  __syncthreads();
  int u = sh_u, v = sh_v;
  const float* zvt = zv_all + (long long)t * N1 * D;
  const float* zgt = zg_all + (long long)t * D;
  int o1 = 1, o2 = 1 + D, o3 = 1 + 2 * D, o4 = 1 + 3 * D;
  int o5 = o4 + A, o6 = o5 + D, o7 = o6 + A;  // P = o7 + D

  // shared part of phi . Wg  (everything not depending on node i)
  float part = 0.0f;
  if (tid < D) {
    int j = tid;
    part = z_pocket[j] * Wg[o1 + j] + z_ligand[j] * Wg[o2 + j] +
           zvt[(long long)u * D + j] * Wg[o3 + j] + zgt[j] * Wg[o7 + j];
    if (j < A) part += xlf[(long long)u * A + j] * Wg[o4 + j];
  }
  red[tid] = part;
  __syncthreads();
  for (int st = 256; st > 0; st >>= 1) {
    if (tid < st) red[tid] += red[tid + st];
    __syncthreads();
  }
  if (tid == 0) sh_c0 = red[0] + (float)t * Wg[0] + bg[0];
  __syncthreads();
  float c0 = sh_c0;

  // per-node score + mask
  float si = -3.0e38f;
  if (tid < N1) {
    int i = tid;
    float d = c0;
    const float* zr = zvt + (long long)i * D;
    for (int j = 0; j < D; ++j) d += zr[j] * Wg[o5 + j];
    for (int j = 0; j < A; ++j) d += xlf[(long long)i * A + j] * Wg[o6 + j];
    float vali = valS[(long long)t * N1 + i];
    float valu = valS[(long long)t * N1 + u];
    float bm = 0.0f;
    if (i == u) bm += NEGV;
    if (vali <= 0.0f) bm += NEGV;
    if (valu <= 0.0f) bm += NEGV;
    bool adj = false;
    for (int s = 0; s < t; ++s)
      if ((eu[s] == u && ev[s] == i) || (ev[s] == u && eu[s] == i)) adj = true;
    if (adj) bm += NEGV;
    float mask = (i == v || i == N1 - 1) ? 0.0f : bm;
    si = d + mask;
    if (i == v) sh_sv = si;
  }
  red[tid] = si;
  __syncthreads();
  for (int st = 256; st > 0; st >>= 1) {
    if (tid < st) red[tid] = fmaxf(red[tid], red[tid + st]);
    __syncthreads();
  }
  float mx = red[0];
  __syncthreads();
  red[tid] = (tid < N1) ? expf(si - mx) : 0.0f;
  __syncthreads();
  for (int st = 256; st > 0; st >>= 1) {
    if (tid < st) red[tid] += red[tid + st];
    __syncthreads();
  }
  if (tid == 0) sh_lse = mx + logf(red[0]);
  __syncthreads();

  // build phi_v and bond logits
  int P = o7 + D;
  for (int idx = tid; idx < P; idx += 512) {
    float val;
    if (idx == 0) val = (float)t;
    else if (idx < o2) val = z_pocket[idx - o1];
    else if (idx < o3) val = z_ligand[idx - o2];
    else if (idx < o4) val = zvt[(long long)u * D + (idx - o3)];
    else if (idx < o5) val = xlf[(long long)u * A + (idx - o4)];
    else if (idx < o6) val = zvt[(long long)v * D + (idx - o5)];
    else if (idx < o7) val = xlf[(long long)v * A + (idx - o6)];
    else val = zgt[idx - o7];
    phiv[idx] = val;
  }
  __syncthreads();
  if (tid < 3) {
    float acc = bh[tid];
    for (int k = 0; k < P; ++k) acc += phiv[k] * Wh[k * 3 + tid];
    float valv = valS[(long long)t * N1 + v];
    if (valv < (float)(tid + 1)) acc += NEGV;
    sh_lg[tid] = acc;
  }
  __syncthreads();
  if (tid == 0) {
    float m3 = fmaxf(sh_lg[0], fmaxf(sh_lg[1], sh_lg[2]));
    float e0 = expf(sh_lg[0] - m3), e1 = expf(sh_lg[1] - m3), e2 = expf(sh_lg[2] - m3);
    float Z = e0 + e1 + e2;
    float pb = (bfs_attr[t * 3 + 0] * e0 + bfs_attr[t * 3 + 1] * e1 +
                bfs_attr[t * 3 + 2] * e2) / Z;
    float lp_v = sh_sv - sh_lse;
    atomAddF(lp_acc, lp_v + logf(pb));
  }
}

// =========================================================== host launcher
extern "C" void kernel_launch(void* const* d_in, const int* in_sizes, int n_in,
                              void* d_out, int out_size, void* d_ws,
                              size_t ws_size, hipStream_t stream) {
  const float* x_p = (const float*)d_in[0];
  const int* eip = (const int*)d_in[1];
  const float* x_l = (const float*)d_in[2];
  const int* eil = (const int*)d_in[3];
  const int* bfs_index = (const int*)d_in[4];
  const float* bfs_attr = (const float*)d_in[5];
  const float* Wp1 = (const float*)d_in[6];
  const float* bp1 = (const float*)d_in[7];
  const float* Wp2 = (const float*)d_in[8];
  const float* bp2 = (const float*)d_in[9];
  const float* Wl1 = (const float*)d_in[10];
  const float* bl1 = (const float*)d_in[11];
  const float* Wl2 = (const float*)d_in[12];
  const float* bl2 = (const float*)d_in[13];
  const float* Wd1 = (const float*)d_in[14];
  const float* bd1 = (const float*)d_in[15];
  const float* Wd2 = (const float*)d_in[16];
  const float* bd2 = (const float*)d_in[17];
  const float* Wf = (const float*)d_in[18];
  const float* bf = (const float*)d_in[19];
  const float* Wg = (const float*)d_in[20];
  const float* bg = (const float*)d_in[21];
  const float* Wh = (const float*)d_in[22];
  const float* bh = (const float*)d_in[23];

  const int D = 128;
  int Fp = in_sizes[6] / D;   // 64
  int Np = in_sizes[0] / Fp;  // 100000
  int Ep = in_sizes[1] / 2;   // 800000
  int Fl = in_sizes[10] / D;  // 16
  int Nl = in_sizes[2] / Fl;  // 400
  int El = in_sizes[3] / 2;   // 798
  int T = in_sizes[4] / 2;    // 399
  int A = in_sizes[18] / D;   // 12
  int N1 = Nl + 1;

  const int* srcP = eip;
  const int* dstP = eip + Ep;
  const int* srcL = eil;
  const int* dstL = eil + El;

  // ---- workspace carving (bump allocator; R1 is reused by scan tensors)
  char* base = (char*)d_ws;
  long long np128 = (long long)Np * D;
  long long scanSz = (long long)T * N1 * D;
  long long r1 = 2 * np128;
  if (scanSz > r1) r1 = scanSz;
  long long off = 0;
  auto alloc = [&](long long nf) -> float* {
    float* p = (float*)(base + off);
    off += nf * 4;
    off = (off + 255) & ~255LL;
    return p;
  };
  float* bufA = alloc(r1);       // pocket h / scan z1,zv (aliased in time)
  float* bufB = bufA + np128;    // pocket agg
  float* z1_all = bufA;
  float* zv_all = bufA;
  float* h2_all = alloc(scanSz);
  float* degP = alloc(Np);
  float* degL = alloc(Nl);
  float* lbufA = alloc((long long)Nl * D);
  float* lbufB = alloc((long long)Nl * D);
  float* zpsum = alloc(D);
  float* zlsum = alloc(D);
  float* z_pocket = alloc(D);
  float* z_ligand = alloc(D);
  float* xlf = alloc((long long)N1 * A);
  float* h1d = alloc((long long)N1 * D);
  float* valS = alloc((long long)T * N1);
  float* degS = alloc((long long)T * N1);
  float* zg_all = alloc((long long)T * D);
  int* us = (int*)alloc(T);
  int* vs = (int*)alloc(T);
  float* lp_acc = alloc(8);

  const int TB = 256;
  // ---------------- pocket GCN (two layers -> mean -> z_pocket)
  fill_kernel<<<256, TB, 0, stream>>>(degP, 1.0f, Np);
  edge_deg_kernel<<<(Ep + TB - 1) / TB, TB, 0, stream>>>(degP, dstP, Ep);
  gemm_wmma_n128<<<dim3((Np + 15) / 16, 1), 32, 0, stream>>>(x_p, Wp1, bufA, Np, Fp, 0, 0);
  fill_kernel<<<2048, TB, 0, stream>>>(bufB, 0.0f, np128);
  edge_agg_kernel<<<(int)(((long long)Ep * 32 + TB - 1) / TB), TB, 0, stream>>>(
      bufB, bufA, degP, srcP, dstP, Ep);
  combine_kernel<<<(int)((np128 + TB - 1) / TB), TB, 0, stream>>>(
      bufB, bufB, bufA, degP, bp1, Np, 1);
  gemm_wmma_n128<<<dim3((Np + 15) / 16, 1), 32, 0, stream>>>(bufB, Wp2, bufA, Np, D, 0, 0);
  fill_kernel<<<2048, TB, 0, stream>>>(bufB, 0.0f, np128);
  edge_agg_kernel<<<(int)(((long long)Ep * 32 + TB - 1) / TB), TB, 0, stream>>>(
      bufB, bufA, degP, srcP, dstP, Ep);
  combine_kernel<<<(int)((np128 + TB - 1) / TB), TB, 0, stream>>>(
      bufB, bufB, bufA, degP, bp2, Np, 0);
  fill_kernel<<<1, 128, 0, stream>>>(zpsum, 0.0f, D);
  rowmean_kernel<<<256, 256, 0, stream>>>(zpsum, bufB, Np);
  vecdiv_kernel<<<1, 128, 0, stream>>>(z_pocket, zpsum, 1.0f / (float)Np);

  // ---------------- ligand GCN -> z_lig_atoms (lbufB), z_ligand
  fill_kernel<<<2, TB, 0, stream>>>(degL, 1.0f, Nl);
  edge_deg_kernel<<<(El + TB - 1) / TB, TB, 0, stream>>>(degL, dstL, El);
  gemm_wmma_n128<<<dim3((Nl + 15) / 16, 1), 32, 0, stream>>>(x_l, Wl1, lbufA, Nl, Fl, 0, 0);
  fill_kernel<<<64, TB, 0, stream>>>(lbufB, 0.0f, (long long)Nl * D);
  edge_agg_kernel<<<(int)(((long long)El * 32 + TB - 1) / TB), TB, 0, stream>>>(
      lbufB, lbufA, degL, srcL, dstL, El);
  combine_kernel<<<(int)(((long long)Nl * D + TB - 1) / TB), TB, 0, stream>>>(
      lbufB, lbufB, lbufA, degL, bl1, Nl, 1);
  gemm_wmma_n128<<<dim3((Nl + 15) / 16, 1), 32, 0, stream>>>(lbufB, Wl2, lbufA, Nl, D, 0, 0);
  fill_kernel<<<64, TB, 0, stream>>>(lbufB, 0.0f, (long long)Nl * D);
  edge_agg_kernel<<<(int)(((long long)El * 32 + TB - 1) / TB), TB, 0, stream>>>(
      lbufB, lbufA, degL, srcL, dstL, El);
  combine_kernel<<<(int)(((long long)Nl * D + TB - 1) / TB), TB, 0, stream>>>(
      lbufB, lbufB, lbufA, degL, bl2, Nl, 0);
  fill_kernel<<<1, 128, 0, stream>>>(zlsum, 0.0f, D);
  rowmean_kernel<<<128, 256, 0, stream>>>(zlsum, lbufB, Nl);
  vecdiv_kernel<<<1, 128, 0, stream>>>(z_ligand, zlsum, 1.0f / (float)Nl);

  // ---------------- labels, base log-prob, shared layer-1 features
  xlabel_kernel<<<(N1 + TB - 1) / TB, TB, 0, stream>>>(xlf, lbufB, Wf, bf, Nl, A);
  fill_kernel<<<1, 32, 0, stream>>>(lp_acc, 0.0f, 1);
  lp0_kernel<<<1, 512, 0, stream>>>(lp_acc, xlf, x_l, Nl, Fl, A);
  gemm_wmma_n128<<<dim3((N1 + 15) / 16, 1), 32, 0, stream>>>(xlf, Wd1, h1d, N1, A, 0, 0);

  // ---------------- deterministic per-step state, then batched scan GCNs
  stepstate_kernel<<<1, 512, 0, stream>>>(us, vs, valS, degS, bfs_index,
                                          bfs_attr, x_l, N1, T, Fl, A);
  z1scan_kernel<<<dim3(N1, T), 128, 0, stream>>>(z1_all, h1d, us, vs, degS, bd1, N1);
  gemm_wmma_n128<<<dim3((N1 + 15) / 16, T), 32, 0, stream>>>(
      z1_all, Wd2, h2_all, N1, D, (long long)N1 * D, (long long)N1 * D);
  zvscan_kernel<<<dim3(N1, T), 128, 0, stream>>>(zv_all, h2_all, us, vs, degS, bd2, N1);
  zg_kernel<<<T, 128, 0, stream>>>(zg_all, zv_all, N1);

  // ---------------- per-step scoring (all steps independent)
  score_kernel<<<T, 512, 0, stream>>>(lp_acc, zv_all, zg_all, xlf, z_pocket,
                                      z_ligand, valS, us, vs, bfs_attr, Wg, bg,
                                      Wh, bh, N1, A);
  writeout_kernel<<<1, 1, 0, stream>>>((float*)d_out, lp_acc);
}